// PiecewiseLinearUnitV1_14516989461089
// MI455X (gfx1250) — compile-verified
//
#include <hip/hip_runtime.h>

namespace {

constexpr int INTERVALS     = 20;     // max(int(N)=2, MAX_N=20) -> static 20
constexpr int TPB           = 256;    // 8 wave32 per block
constexpr int V4_PER_THREAD = 4;      // 16 floats / thread -> 4x b128 load clause
constexpr long long PF_DIST_V4 = 1 << 17;  // prefetch ~2 MB ahead (131072 float4)

// Native clang vector types (accepted by nontemporal builtins; b128/b64 ops).
typedef float v4f __attribute__((ext_vector_type(4)));
typedef float v2f __attribute__((ext_vector_type(2)));

// One ds_load_b64 gather fetches (Y0, Kidx) together.
__device__ __forceinline__ float plu_eval(float xi,
                                          float Bl, float Br,
                                          float Kl, float Kr,
                                          float il, bool il_nz,
                                          float h0, float htop,
                                          const v2f* __restrict__ tab) {
    // idx = floor(div_no_nan(x - Bl, il)) as int32; true fdiv -> bit-exact
    // index selection vs the reference.
    float q  = il_nz ? (xi - Bl) / il : 0.0f;
    int  idx = (int)floorf(q);
    int  i0  = min(max(idx, 0), INTERVALS);   // take(..., mode="clip")
    v2f  yk  = tab[i0];                       // ds_load_b64: {Y0, Kidx}
    float y0   = yk.x;
    float kidx = yk.y;   // ==(nh[i0+1]-nh[i0])/il for i0<20; 0 when clipped pair
                         // collapses (those terms are masked by exact b3=0.0)
    // masks as 0.0/1.0 floats (NaN input -> all masks 0 -> 0*NaN = NaN, as ref)
    float b1 = (xi <  Bl) ? 1.0f : 0.0f;
    float b2 = (xi >= Br) ? 1.0f : 0.0f;
    float b3 = ((xi >= Bl) && (xi < Br)) ? 1.0f : 0.0f;
    float l1 = (xi - Bl) * Kl + h0;
    float l2 = (xi - Br) * Kr + htop;            // top = nheight[20]
    float l3 = (xi - floorf(xi)) * kidx + y0;    // Bidx = floor(x)
    return b1 * l1 + b2 * l2 + b3 * l3;
}

__global__ __launch_bounds__(TPB) void plu_kernel(
    const float* __restrict__ x,
    const float* __restrict__ bounds,
    const float* __restrict__ bslope,
    const float* __restrict__ nh,
    float* __restrict__ out,
    long long n4,      // number of float4 chunks
    int tail)          // n - 4*n4 leftover scalars
{
    // Uniform scalar parameters (s_load path: uniform pointer + const offset).
    const float Bl = bounds[0];
    const float Br = bounds[1];
    const float Kl = bslope[0];
    const float Kr = bslope[1];
    const float il = (Br - Bl) * (1.0f / (float)INTERVALS);
    const bool  il_nz = (il != 0.0f);
    const float h0   = nh[0];            // uniform s_load
    const float htop = nh[INTERVALS];    // top = nheight[20]

    // Stage paired (height, slope) table in LDS: one b64 gather per element.
    __shared__ v2f tab[INTERVALS + 1];
    if (threadIdx.x <= INTERVALS) {
        int i  = (int)threadIdx.x;
        float y0 = nh[i];
        float y1 = nh[min(i + 1, INTERVALS)];
        // Same fdiv the reference performs -> bit-identical slope for the
        // in-range intervals; exactly 0 when the clipped pair collapses or
        // when interval_length == 0 (div_no_nan).
        float k  = il_nz ? (y1 - y0) / il : 0.0f;
        v2f p; p.x = y0; p.y = k;
        tab[i] = p;
    }
    __syncthreads();

    const v4f* __restrict__ x4 = reinterpret_cast<const v4f*>(x);
    v4f* __restrict__       o4 = reinterpret_cast<v4f*>(out);

    const long long perBlock  = (long long)TPB * V4_PER_THREAD;
    const long long blockBase = (long long)blockIdx.x * perBlock;
    const long long base      = blockBase + threadIdx.x;

    const bool fullBlock = (blockBase + perBlock) <= n4;

    if (fullBlock) {
        // Deep prefetch for trailing blocks: gfx1250 global_prefetch_b8 into L2.
        if (base + PF_DIST_V4 < n4) {
            __builtin_prefetch(&x4[base + PF_DIST_V4], 0, 1);
        }
        // 4 back-to-back non-temporal b128 loads (streaming: don't pollute L2).
        v4f v[V4_PER_THREAD];
#pragma unroll
        for (int k = 0; k < V4_PER_THREAD; ++k)
            v[k] = __builtin_nontemporal_load(&x4[base + (long long)k * TPB]);
#pragma unroll
        for (int k = 0; k < V4_PER_THREAD; ++k) {
            v4f r;
#pragma unroll
            for (int j = 0; j < 4; ++j)
                r[j] = plu_eval(v[k][j], Bl, Br, Kl, Kr, il, il_nz, h0, htop, tab);
            __builtin_nontemporal_store(r, &o4[base + (long long)k * TPB]);
        }
    } else {
        // Ragged last block: per-chunk guard.
#pragma unroll
        for (int k = 0; k < V4_PER_THREAD; ++k) {
            long long i = base + (long long)k * TPB;
            if (i < n4) {
                v4f v = __builtin_nontemporal_load(&x4[i]);
                v4f r;
#pragma unroll
                for (int j = 0; j < 4; ++j)
                    r[j] = plu_eval(v[j], Bl, Br, Kl, Kr, il, il_nz, h0, htop, tab);
                __builtin_nontemporal_store(r, &o4[i]);
            }
        }
    }

    // Scalar tail (n not divisible by 4) handled by block 0.
    if (blockIdx.x == 0 && (int)threadIdx.x < tail) {
        long long i = n4 * 4 + threadIdx.x;
        float xi = x[i];
        out[i] = plu_eval(xi, Bl, Br, Kl, Kr, il, il_nz, h0, htop, tab);
    }
}

} // anonymous namespace

extern "C" void kernel_launch(void* const* d_in, const int* in_sizes, int n_in,
                              void* d_out, int out_size, void* d_ws, size_t ws_size,
                              hipStream_t stream) {
    // setup_inputs() order: inputs, N, Bounds, BoundSlope, nheight
    const float* x      = (const float*)d_in[0];
    // d_in[1] = N: unused, intervals is the static constant 20 in the reference
    const float* bounds = (const float*)d_in[2];
    const float* bslope = (const float*)d_in[3];
    const float* nh     = (const float*)d_in[4];
    float* out          = (float*)d_out;

    const long long n  = (long long)out_size;   // 32*4096*768 = 100,663,296
    const long long n4 = n / 4;
    const int tail     = (int)(n - n4 * 4);

    const long long perBlock = (long long)TPB * V4_PER_THREAD;   // 1024 float4 / block
    long long nblocks = (n4 + perBlock - 1) / perBlock;          // 24,576 blocks
    if (nblocks == 0) nblocks = 1;                               // still run tail path

    plu_kernel<<<(int)nblocks, TPB, 0, stream>>>(x, bounds, bslope, nh, out, n4, tail);
}